// WKVOperator_37374805409856
// MI455X (gfx1250) — compile-verified
//
#include <hip/hip_runtime.h>
#include <stdint.h>

// Problem constants (match reference).
#define HID   2048
#define BATCH 4
#define TLEN  4096
#define BH    (BATCH * HID)     // 8192 channels
#define NC    64                // time chunks
#define CT    (TLEN / NC)       // 64 steps per chunk
#define BLK   128               // threads per block (4 wave32s)
#define G4    (BH / 4)          // thread-columns (4 channels per thread)
#define DEPTH 8                 // async LDS prefetch depth (k,v pairs in flight)
#define EPS   1e-8f

// ---------------------------------------------------------------------------
// CDNA5 async global -> LDS copy (ASYNCcnt-tracked). Each active lane copies
// 16B from its own global address into its own LDS byte offset.
// GV mode: global_load_async_to_lds_b128 vdst(LDS addr), vaddr(64b), off
// ---------------------------------------------------------------------------
__device__ __forceinline__ void async_load16(uint32_t lds_off, const float* g) {
  asm volatile("global_load_async_to_lds_b128 %0, %1, off"
               :: "v"(lds_off), "v"((unsigned long long)(uintptr_t)g)
               : "memory");
}

// Refill a ring slot: the s_wait_dscnt 0 guarantees the ds_load reads of this
// slot have architecturally completed before the overwriting async load can
// issue (closes the WAR window the compiler's own waits don't cover).
__device__ __forceinline__ void async_refill_pair(uint32_t klds, const float* kg,
                                                  uint32_t vlds, const float* vg) {
  asm volatile("s_wait_dscnt 0x0\n\t"
               "global_load_async_to_lds_b128 %0, %1, off\n\t"
               "global_load_async_to_lds_b128 %2, %3, off"
               :: "v"(klds), "v"((unsigned long long)(uintptr_t)kg),
                  "v"(vlds), "v"((unsigned long long)(uintptr_t)vg)
               : "memory");
}

// ---------------------------------------------------------------------------
// Pass 1: chunk-local scan sums. Thread = (4 channels, 1 chunk).
// Runs the reference recurrence within the chunk starting from 0, yielding
// A_j = sum_{t in chunk} ew^{(tend-1-t)} e^{k_t} v_t  (B_j likewise, v=1).
// ---------------------------------------------------------------------------
__global__ void __launch_bounds__(BLK) wkv_pass1(
    const float* __restrict__ k, const float* __restrict__ v,
    const float* __restrict__ td,
    float* __restrict__ sA, float* __restrict__ sB)
{
  const int g  = blockIdx.x * BLK + threadIdx.x;   // [0, G4)
  const int h4 = g * 4;
  const int bb = h4 / HID;
  const int h  = h4 % HID;
  const int j  = blockIdx.y;                       // chunk

  const float4 tdq = *(const float4*)&td[h];
  const float ew0 = __expf(-__expf(tdq.x));
  const float ew1 = __expf(-__expf(tdq.y));
  const float ew2 = __expf(-__expf(tdq.z));
  const float ew3 = __expf(-__expf(tdq.w));

  float a0 = 0.f, a1 = 0.f, a2 = 0.f, a3 = 0.f;
  float s0 = 0.f, s1 = 0.f, s2 = 0.f, s3 = 0.f;

  const size_t base = (size_t)bb * TLEN * HID + (size_t)j * CT * HID + h;
  const float* kp = k + base;
  const float* vp = v + base;

#pragma unroll 4
  for (int t = 0; t < CT; ++t) {
    const float4 kq = *(const float4*)kp; kp += HID;
    const float4 vq = *(const float4*)vp; vp += HID;
    const float e0 = __expf(kq.x), e1 = __expf(kq.y);
    const float e2 = __expf(kq.z), e3 = __expf(kq.w);
    a0 = fmaf(ew0, a0, e0 * vq.x); s0 = fmaf(ew0, s0, e0);
    a1 = fmaf(ew1, a1, e1 * vq.y); s1 = fmaf(ew1, s1, e1);
    a2 = fmaf(ew2, a2, e2 * vq.z); s2 = fmaf(ew2, s2, e2);
    a3 = fmaf(ew3, a3, e3 * vq.w); s3 = fmaf(ew3, s3, e3);
  }

  const int o = j * BH + h4;
  *(float4*)&sA[o] = make_float4(a0, a1, a2, a3);
  *(float4*)&sB[o] = make_float4(s0, s1, s2, s3);
}

// ---------------------------------------------------------------------------
// Pass 2: sequential prefix across the NC chunk states per channel.
// Combine: state_{j+1} = ew^CT * state_j + Sum_j.  Writes exclusive-prefix
// carry-in states cA/cB and the final a,b outputs.
// ---------------------------------------------------------------------------
__global__ void __launch_bounds__(BLK) wkv_pass2(
    const float* __restrict__ td,
    const float* __restrict__ sA, const float* __restrict__ sB,
    float* __restrict__ cA, float* __restrict__ cB,
    float* __restrict__ aout, float* __restrict__ bout)
{
  const int g  = blockIdx.x * BLK + threadIdx.x;   // [0, G4)
  const int h4 = g * 4;
  const int h  = h4 % HID;

  const float4 tdq = *(const float4*)&td[h];
  float p0 = __expf(-__expf(tdq.x));
  float p1 = __expf(-__expf(tdq.y));
  float p2 = __expf(-__expf(tdq.z));
  float p3 = __expf(-__expf(tdq.w));
  // ew^CT, CT = 64 = 2^6 -> 6 squarings
#pragma unroll
  for (int s = 0; s < 6; ++s) { p0 *= p0; p1 *= p1; p2 *= p2; p3 *= p3; }

  float a0 = 0.f, a1 = 0.f, a2 = 0.f, a3 = 0.f;
  float b0 = 0.f, b1 = 0.f, b2 = 0.f, b3 = 0.f;

#pragma unroll 4
  for (int j = 0; j < NC; ++j) {
    const int o = j * BH + h4;
    *(float4*)&cA[o] = make_float4(a0, a1, a2, a3);
    *(float4*)&cB[o] = make_float4(b0, b1, b2, b3);
    const float4 A  = *(const float4*)&sA[o];
    const float4 Bq = *(const float4*)&sB[o];
    a0 = fmaf(p0, a0, A.x);  b0 = fmaf(p0, b0, Bq.x);
    a1 = fmaf(p1, a1, A.y);  b1 = fmaf(p1, b1, Bq.y);
    a2 = fmaf(p2, a2, A.z);  b2 = fmaf(p2, b2, Bq.z);
    a3 = fmaf(p3, a3, A.w);  b3 = fmaf(p3, b3, Bq.w);
  }

  *(float4*)&aout[h4] = make_float4(a0, a1, a2, a3);
  *(float4*)&bout[h4] = make_float4(b0, b1, b2, b3);
}

// ---------------------------------------------------------------------------
// Pass 3: emit wkv outputs. Streams k,v through a DEPTH-deep LDS ring via
// CDNA5 async global->LDS loads (ASYNCcnt-gated software pipeline). Each lane
// reads only the 16B it copied, so no barriers are needed. Constant issue
// rate (clamped tail) keeps the wait immediate fixed at 2*(DEPTH-1)=14.
// ---------------------------------------------------------------------------
__global__ void __launch_bounds__(BLK) wkv_pass3(
    const float* __restrict__ k, const float* __restrict__ v,
    const float* __restrict__ td, const float* __restrict__ tf,
    const float* __restrict__ cA, const float* __restrict__ cB,
    float* __restrict__ out)
{
  __shared__ float kbuf[DEPTH * BLK * 4];   // 16 KB
  __shared__ float vbuf[DEPTH * BLK * 4];   // 16 KB

  const int g  = blockIdx.x * BLK + threadIdx.x;
  const int h4 = g * 4;
  const int bb = h4 / HID;
  const int h  = h4 % HID;
  const int j  = blockIdx.y;

  const float4 tdq = *(const float4*)&td[h];
  const float ew0 = __expf(-__expf(tdq.x));
  const float ew1 = __expf(-__expf(tdq.y));
  const float ew2 = __expf(-__expf(tdq.z));
  const float ew3 = __expf(-__expf(tdq.w));
  const float4 uq = *(const float4*)&tf[h];

  const int o = j * BH + h4;
  const float4 aq = *(const float4*)&cA[o];
  const float4 bq = *(const float4*)&cB[o];
  float a0 = aq.x, a1 = aq.y, a2 = aq.z, a3 = aq.w;
  float b0 = bq.x, b1 = bq.y, b2 = bq.z, b3 = bq.w;

  const size_t base = (size_t)bb * TLEN * HID + (size_t)j * CT * HID + h;
  const float* kp = k + base;
  const float* vp = v + base;
  float*       op = out + base;

  const uint32_t klds  = (uint32_t)(uintptr_t)(&kbuf[0]) + (uint32_t)threadIdx.x * 16u;
  const uint32_t vlds  = (uint32_t)(uintptr_t)(&vbuf[0]) + (uint32_t)threadIdx.x * 16u;
  const uint32_t slotB = (uint32_t)BLK * 16u;   // bytes per ring slot (2 KB)

  // Prologue: fill the ring (DEPTH (k,v) pairs in flight).
#pragma unroll
  for (int i = 0; i < DEPTH; ++i) {
    async_load16(klds + (uint32_t)i * slotB, kp + (size_t)i * HID);
    async_load16(vlds + (uint32_t)i * slotB, vp + (size_t)i * HID);
  }

#pragma unroll 2
  for (int t = 0; t < CT; ++t) {
    // Oldest pair complete when <= 2*(DEPTH-1) async ops remain outstanding
    // (async loads complete in order).
    asm volatile("s_wait_asynccnt 14" ::: "memory");
    const int slot = t & (DEPTH - 1);
    const int ro   = slot * BLK * 4 + (int)threadIdx.x * 4;
    const float4 kq = *(const float4*)&kbuf[ro];
    const float4 vq = *(const float4*)&vbuf[ro];

    // wkv_t uses the state BEFORE the update (exclusive prefix).
    const float eu0 = __expf(uq.x + kq.x);
    const float eu1 = __expf(uq.y + kq.y);
    const float eu2 = __expf(uq.z + kq.z);
    const float eu3 = __expf(uq.w + kq.w);
    const float w0 = __fdividef(fmaf(eu0, vq.x, a0), b0 + eu0 + EPS);
    const float w1 = __fdividef(fmaf(eu1, vq.y, a1), b1 + eu1 + EPS);
    const float w2 = __fdividef(fmaf(eu2, vq.z, a2), b2 + eu2 + EPS);
    const float w3 = __fdividef(fmaf(eu3, vq.w, a3), b3 + eu3 + EPS);
    *(float4*)op = make_float4(w0, w1, w2, w3);
    op += HID;

    const float e0 = __expf(kq.x), e1 = __expf(kq.y);
    const float e2 = __expf(kq.z), e3 = __expf(kq.w);
    a0 = fmaf(ew0, a0, e0 * vq.x); b0 = fmaf(ew0, b0, e0);
    a1 = fmaf(ew1, a1, e1 * vq.y); b1 = fmaf(ew1, b1, e1);
    a2 = fmaf(ew2, a2, e2 * vq.z); b2 = fmaf(ew2, b2, e2);
    a3 = fmaf(ew3, a3, e3 * vq.w); b3 = fmaf(ew3, b3, e3);

    // Refill this slot for t+DEPTH (clamped dummy issues at the tail keep the
    // ASYNCcnt arithmetic uniform; dummy slots are never read). The embedded
    // s_wait_dscnt 0 orders the refill after this slot's ds_load reads.
    int tn = t + DEPTH; tn = (tn < CT) ? tn : (CT - 1);
    async_refill_pair(klds + (uint32_t)slot * slotB, kp + (size_t)tn * HID,
                      vlds + (uint32_t)slot * slotB, vp + (size_t)tn * HID);
  }
}

// ---------------------------------------------------------------------------
// Launch: d_in = {k, v, time_decay, time_first}; d_out = [wkv | a | b].
// Workspace: 4 arrays of BH*NC floats (8 MiB total with NC=64).
// ---------------------------------------------------------------------------
extern "C" void kernel_launch(void* const* d_in, const int* in_sizes, int n_in,
                              void* d_out, int out_size, void* d_ws, size_t ws_size,
                              hipStream_t stream)
{
  (void)in_sizes; (void)n_in; (void)out_size; (void)ws_size;

  const float* k  = (const float*)d_in[0];
  const float* v  = (const float*)d_in[1];
  const float* td = (const float*)d_in[2];
  const float* tf = (const float*)d_in[3];

  float* out  = (float*)d_out;
  float* aout = out + (size_t)BATCH * TLEN * HID;
  float* bout = aout + BH;

  float* sA = (float*)d_ws;
  float* sB = sA + (size_t)BH * NC;
  float* cA = sB + (size_t)BH * NC;
  float* cB = cA + (size_t)BH * NC;

  dim3 blk(BLK);
  dim3 grid1(G4 / BLK, NC);   // (16, 64) blocks
  wkv_pass1<<<grid1, blk, 0, stream>>>(k, v, td, sA, sB);
  wkv_pass2<<<dim3(G4 / BLK), blk, 0, stream>>>(td, sA, sB, cA, cB, aout, bout);
  wkv_pass3<<<grid1, blk, 0, stream>>>(k, v, td, tf, cA, cB, out);
}